// DecoderRNN_59004260712896
// MI455X (gfx1250) — compile-verified
//
#include <hip/hip_runtime.h>

#define B_ 64
#define T_ 21
#define E_ 512
#define H_ 512
#define G_ 2048   // 4*H
#define V_ 10000

typedef float v2f __attribute__((ext_vector_type(2)));
typedef float v8f __attribute__((ext_vector_type(8)));

__device__ __forceinline__ v8f wmma4(v2f a, v2f b, v8f c) {
  // V_WMMA_F32_16X16X4_F32 : D = A(16x4) * B(4x16) + C
  return __builtin_amdgcn_wmma_f32_16x16x4_f32(false, a, false, b, (short)0, c,
                                               false, false);
}

__device__ __forceinline__ v2f ld2(const float* p) { return *(const v2f*)p; }

__device__ __forceinline__ float sigmf(float x) {
  return 1.0f / (1.0f + __expf(-x));
}

// ---------------------------------------------------------------- init h0, c
__global__ void k_init_zero(float* __restrict__ h0, float* __restrict__ c) {
  int i = blockIdx.x * blockDim.x + threadIdx.x;
  if (i < B_ * H_) { h0[i] = 0.0f; c[i] = 0.0f; }
}

// ------------------------------------------- Xp[m,g] = x[m]@W_ih^T + (b_ih+b_hh)
// M = T*B = 1344 rows (m = t*64+b), N = 2048, K = 512.
// Wave = 16 rows x 64 cols (4 N-tiles). Waves of a block share n0 (B reuse
// via L0), differ in mt. 21 M-groups x 32 N-groups = 672 blocks.
__global__ void k_xproj(const float* __restrict__ features,
                        const float* __restrict__ cap,     // [B,T,E]
                        const float* __restrict__ W_ih,    // [4H,E]
                        const float* __restrict__ b_ih,
                        const float* __restrict__ b_hh,
                        float* __restrict__ Xp) {          // [T*B, 4H]
  const int lane = threadIdx.x & 31;
  const int wave = threadIdx.x >> 5;
  const int mg   = blockIdx.x % 21;         // M group
  const int ng   = blockIdx.x / 21;         // 0..31
  const int mt   = mg * 4 + wave;           // 0..83
  const int n0   = ng * 64;
  const int l15  = lane & 15;
  const int half = lane >> 4;
  const int koff = half * 2;

  const int m = mt * 16 + l15;
  const int t = m / B_;
  const int b = m % B_;
  const float* arow = (t == 0) ? (features + (size_t)b * E_)
                               : (cap + ((size_t)b * T_ + (t - 1)) * E_);
  const float* br0 = W_ih + (size_t)(n0 +  0 + l15) * E_;
  const float* br1 = W_ih + (size_t)(n0 + 16 + l15) * E_;
  const float* br2 = W_ih + (size_t)(n0 + 32 + l15) * E_;
  const float* br3 = W_ih + (size_t)(n0 + 48 + l15) * E_;

  v8f a0 = {}, a1 = {}, a2 = {}, a3 = {};
#pragma unroll 8
  for (int kk = 0; kk < E_; kk += 4) {
    v2f a = ld2(arow + kk + koff);
    a0 = wmma4(a, ld2(br0 + kk + koff), a0);
    a1 = wmma4(a, ld2(br1 + kk + koff), a1);
    a2 = wmma4(a, ld2(br2 + kk + koff), a2);
    a3 = wmma4(a, ld2(br3 + kk + koff), a3);
  }
#pragma unroll
  for (int j = 0; j < 4; ++j) {
    v8f acc = (j == 0) ? a0 : (j == 1) ? a1 : (j == 2) ? a2 : a3;
    int n = n0 + j * 16 + l15;
    float bias = b_ih[n] + b_hh[n];
#pragma unroll
    for (int r = 0; r < 8; ++r) {
      int mr = mt * 16 + r + 8 * half;
      Xp[(size_t)mr * G_ + n] = acc[r] + bias;
    }
  }
}

// -------- one LSTM step: gates = Xp_t + h@W_hh^T, fused cell update.
// Wave = 16(batch) x 16(hidden) computing all 4 gate tiles -> owns update.
// Block = 4 waves = all 4 M-tiles sharing one hid-tile (shared W_hh loads).
// 32 blocks (one per hid-tile).
__global__ void k_lstm_step(int t,
                            const float* __restrict__ xp_t,  // [B, 4H]
                            const float* __restrict__ W_hh,  // [4H, H]
                            const float* __restrict__ h_in,  // [B, H]
                            float* __restrict__ h_out,       // [B, H]
                            float* __restrict__ c,           // [B, H]
                            float* __restrict__ Hs) {        // [T-1, B, H]
  const int lane = threadIdx.x & 31;
  const int mt   = threadIdx.x >> 5;        // 0..3 (wave = M-tile)
  const int hid0 = blockIdx.x * 16;         // 0..511
  const int l15  = lane & 15;
  const int half = lane >> 4;
  const int koff = half * 2;

  const float* arow = h_in + (size_t)(mt * 16 + l15) * H_;
  const float* bi = W_hh + (size_t)(0 * H_ + hid0 + l15) * H_;
  const float* bf = W_hh + (size_t)(1 * H_ + hid0 + l15) * H_;
  const float* bg = W_hh + (size_t)(2 * H_ + hid0 + l15) * H_;
  const float* bo = W_hh + (size_t)(3 * H_ + hid0 + l15) * H_;

  v8f ai = {}, af = {}, ag = {}, ao = {};
#pragma unroll 8
  for (int kk = 0; kk < H_; kk += 4) {
    v2f a = ld2(arow + kk + koff);
    ai = wmma4(a, ld2(bi + kk + koff), ai);
    af = wmma4(a, ld2(bf + kk + koff), af);
    ag = wmma4(a, ld2(bg + kk + koff), ag);
    ao = wmma4(a, ld2(bo + kk + koff), ao);
  }
  const int hd = hid0 + l15;
#pragma unroll
  for (int r = 0; r < 8; ++r) {
    int m = mt * 16 + r + 8 * half;            // batch row
    const float* xr = xp_t + (size_t)m * G_;
    float gi = sigmf(ai[r] + xr[hd]);
    float gf = sigmf(af[r] + xr[H_ + hd]);
    float gg = tanhf (ag[r] + xr[2 * H_ + hd]);
    float go = sigmf(ao[r] + xr[3 * H_ + hd]);
    size_t ci = (size_t)m * H_ + hd;
    float cn = gf * c[ci] + gi * gg;
    c[ci] = cn;
    float hn = go * tanhf(cn);
    h_out[ci] = hn;
    if (t >= 1) Hs[((size_t)(t - 1) * B_ + m) * H_ + hd] = hn;
  }
}

// -------- vocab projection: Out[b,t,v] = Hs[t,b]@W_lin^T + b_lin
// M = 20*64 = 1280 (m = t*64+b), N = 10000 = 625 tiles, K = 512.
// Wave = 2 M-tiles (32 rows) x 80 cols (5 N-tiles): B fragment amortized
// over 2 WMMAs. Block = 4 waves with consecutive M-pairs sharing n0.
// 10 M-pair-groups x 125 N-groups = 1250 blocks.
__global__ void k_vocab(const float* __restrict__ Hs,     // [1280, 512]
                        const float* __restrict__ W_lin,  // [V, H]
                        const float* __restrict__ b_lin,
                        float* __restrict__ out) {        // [B, T-1, V]
  const int lane = threadIdx.x & 31;
  const int wave = threadIdx.x >> 5;
  const int pg   = blockIdx.x % 10;          // M-pair group
  const int ngp  = blockIdx.x / 10;          // 0..124
  const int p    = pg * 4 + wave;            // 0..39 (rows 32p..32p+31)
  const int n0   = ngp * 80;
  const int l15  = lane & 15;
  const int half = lane >> 4;
  const int koff = half * 2;

  const float* arow0 = Hs + (size_t)(p * 32      + l15) * H_;
  const float* arow1 = Hs + (size_t)(p * 32 + 16 + l15) * H_;
  const float* brow[5];
#pragma unroll
  for (int j = 0; j < 5; ++j)
    brow[j] = W_lin + (size_t)(n0 + j * 16 + l15) * H_;

  v8f acc0[5] = {};
  v8f acc1[5] = {};
#pragma unroll 2
  for (int kk = 0; kk < H_; kk += 4) {
    v2f a0 = ld2(arow0 + kk + koff);
    v2f a1 = ld2(arow1 + kk + koff);
#pragma unroll
    for (int j = 0; j < 5; ++j) {
      v2f b = ld2(brow[j] + kk + koff);
      acc0[j] = wmma4(a0, b, acc0[j]);
      acc1[j] = wmma4(a1, b, acc1[j]);
    }
  }
#pragma unroll
  for (int j = 0; j < 5; ++j) {
    int n = n0 + j * 16 + l15;
    float bias = b_lin[n];
#pragma unroll
    for (int r = 0; r < 8; ++r) {
      {
        int m  = p * 32 + r + 8 * half;
        int b  = m % B_;
        int tt = m / B_;
        out[((size_t)b * (T_ - 1) + tt) * V_ + n] = acc0[j][r] + bias;
      }
      {
        int m  = p * 32 + 16 + r + 8 * half;
        int b  = m % B_;
        int tt = m / B_;
        out[((size_t)b * (T_ - 1) + tt) * V_ + n] = acc1[j][r] + bias;
      }
    }
  }
}

extern "C" void kernel_launch(void* const* d_in, const int* in_sizes, int n_in,
                              void* d_out, int out_size, void* d_ws, size_t ws_size,
                              hipStream_t stream) {
  const float* features = (const float*)d_in[0];
  const float* cap      = (const float*)d_in[1];
  const float* W_ih     = (const float*)d_in[2];
  const float* W_hh     = (const float*)d_in[3];
  const float* b_ih     = (const float*)d_in[4];
  const float* b_hh     = (const float*)d_in[5];
  const float* W_lin    = (const float*)d_in[6];
  const float* b_lin    = (const float*)d_in[7];
  float* out = (float*)d_out;

  float* Xp = (float*)d_ws;                       // T*B*4H
  float* h0 = Xp + (size_t)T_ * B_ * G_;          // B*H
  float* h1 = h0 + (size_t)B_ * H_;               // B*H
  float* c  = h1 + (size_t)B_ * H_;               // B*H
  float* Hs = c  + (size_t)B_ * H_;               // (T-1)*B*H

  k_init_zero<<<(B_ * H_ + 255) / 256, 256, 0, stream>>>(h0, c);
  k_xproj<<<672, 128, 0, stream>>>(features, cap, W_ih, b_ih, b_hh, Xp);
  for (int t = 0; t < T_; ++t) {
    float* hin  = (t & 1) ? h1 : h0;
    float* hout = (t & 1) ? h0 : h1;
    k_lstm_step<<<32, 128, 0, stream>>>(t, Xp + (size_t)t * B_ * G_, W_hh,
                                        hin, hout, c, Hs);
  }
  k_vocab<<<1250, 128, 0, stream>>>(Hs, W_lin, b_lin, out);
}